// GarNet_9345848836679
// MI455X (gfx1250) — compile-verified
//
#include <hip/hip_runtime.h>
#include <math.h>

typedef __attribute__((ext_vector_type(2))) float v2f;
typedef __attribute__((ext_vector_type(8))) float v8f;

// Shapes
#define BB 64
#define TT 2048
#define NF 16
#define FD 64
#define AA 8

// ---------------------------------------------------------------------------
// Async global -> LDS staging of a 64 x CIN fp32 tile (row stride SRC_STRIDE
// floats in LDS), using GLOBAL_LOAD_ASYNC_TO_LDS_B128 (ASYNCcnt-tracked).
// Issue-only: caller performs s_wait_asynccnt + barrier.
// Per-wave instruction count for one tile = CIN/16 (OPS/256 iterations).
// ---------------------------------------------------------------------------
__device__ inline unsigned lds_off(const void* p) {
  // Low 32 bits of a flat shared-aperture address are the LDS byte offset.
  return (unsigned)(unsigned long long)p;
}

template <int CIN, int SRC_STRIDE>
__device__ inline void stage_async_issue(const float* __restrict__ gsrc,
                                         float* __restrict__ sdst, int tid) {
  constexpr int OPS = 64 * CIN / 4;  // number of b128 transfers
  const unsigned sbase = lds_off(sdst);
#pragma unroll
  for (int i = tid; i < OPS; i += 256) {
    const int r = i / (CIN / 4);
    const int c4 = i % (CIN / 4);
    unsigned ldsoff = sbase + (unsigned)(r * SRC_STRIDE + c4 * 4) * 4u;
    unsigned goff = (unsigned)(r * CIN + c4 * 4) * 4u;
    asm volatile("global_load_async_to_lds_b128 %0, %1, %2 offset:0"
                 :
                 : "v"(ldsoff), "v"(goff), "s"(gsrc)
                 : "memory");
  }
}

template <int N>
__device__ inline void wait_asynccnt() {
  asm volatile("s_wait_asynccnt %0" : : "n"(N) : "memory");
}

// ---------------------------------------------------------------------------
// Cooperative 8-wave WMMA GEMM: OUT[64][NOUT] = SRC[64][CIN] @ W[CIN][NOUT]
// All operands in LDS. V_WMMA_F32_16X16X4_F32.
// A frag: row = lane%16, K = (lane/16)*2 + v   (2 VGPRs)
// B frag: col = lane%16, K = (lane/16)*2 + v   (2 VGPRs)
// D frag: M = v + 8*(lane/16), N = lane%16     (8 VGPRs)
// ---------------------------------------------------------------------------
template <int CIN, int SRC_STRIDE, int NOUT, int OUT_STRIDE>
__device__ inline void wmma_gemm64(const float* __restrict__ sA,
                                   const float* __restrict__ sW,
                                   float* __restrict__ sO) {
  const int lane  = threadIdx.x & 31;
  const int wave  = __builtin_amdgcn_readfirstlane(threadIdx.x >> 5);  // 0..7
  const int lhalf = lane >> 4;
  const int lmod  = lane & 15;
  constexpr int NT = NOUT / 16;
  for (int tile = wave; tile < 4 * NT; tile += 8) {
    const int mt = tile / NT;
    const int nt = tile % NT;
    v8f acc = {};
#pragma unroll
    for (int k0 = 0; k0 < CIN; k0 += 4) {
      const int kk = k0 + lhalf * 2;
      v2f a, b;
      a[0] = sA[(mt * 16 + lmod) * SRC_STRIDE + kk];
      a[1] = sA[(mt * 16 + lmod) * SRC_STRIDE + kk + 1];
      b[0] = sW[kk * NOUT + nt * 16 + lmod];
      b[1] = sW[(kk + 1) * NOUT + nt * 16 + lmod];
      acc = __builtin_amdgcn_wmma_f32_16x16x4_f32(false, a, false, b,
                                                  (short)0, acc, false, false);
    }
#pragma unroll
    for (int v = 0; v < 8; ++v)
      sO[(mt * 16 + v + 8 * lhalf) * OUT_STRIDE + nt * 16 + lmod] = acc[v];
  }
}

// ---------------------------------------------------------------------------
// pool_kernel: per batch, max/mean over T of edges f*exp(-|d|),
// [f|d] = src @ w_in + b_in.  pooled[b][a*128 + fd] = max, +64+fd = mean.
// Double-buffered async LDS staging over 32 T-chunks.
// ---------------------------------------------------------------------------
template <int CIN>
__global__ __launch_bounds__(256) void pool_kernel(
    const float* __restrict__ src,   // B x T x CIN
    const float* __restrict__ w_in,  // CIN x 72
    const float* __restrict__ b_in,  // 72
    float* __restrict__ pooled) {    // B x 1024
  constexpr int SRC_STRIDE = CIN + 4;
  constexpr int XPS = 84;            // 80 cols padded
  constexpr int OPW = CIN / 16;      // async b128 ops per wave per chunk
  __shared__ __align__(16) float sW[CIN * 80];
  __shared__ float sB[80];
  __shared__ __align__(16) float sSrc0[64 * SRC_STRIDE];
  __shared__ __align__(16) float sSrc1[64 * SRC_STRIDE];
  __shared__ __align__(16) float sXP[64 * XPS];

  const int b = blockIdx.x;
  const int tid = threadIdx.x;
  const float* gsrc = src + (size_t)b * TT * CIN;

  // prefetch chunk 0 while staging weights
  stage_async_issue<CIN, SRC_STRIDE>(gsrc, sSrc0, tid);

  for (int i = tid; i < CIN * 80; i += 256) {
    int r = i / 80, c = i % 80;
    sW[i] = (c < 72) ? w_in[r * 72 + c] : 0.f;
  }
  if (tid < 80) sB[tid] = (tid < 72) ? b_in[tid] : 0.f;

  const int a = tid >> 5;          // 0..7
  const int fd0 = (tid & 31) * 2;  // 0,2,..,62
  float mx0 = -3.4e38f, mx1 = -3.4e38f, sm0 = 0.f, sm1 = 0.f;

  int cur = 0;
  for (int tc = 0; tc < TT; tc += 64) {
    const float* curbuf = cur ? sSrc1 : sSrc0;
    float* nxtbuf = cur ? sSrc0 : sSrc1;
    if (tc + 64 < TT) {
      stage_async_issue<CIN, SRC_STRIDE>(gsrc + (size_t)(tc + 64) * CIN,
                                         nxtbuf, tid);
      wait_asynccnt<OPW>();  // in-order: current chunk has landed
    } else {
      wait_asynccnt<0>();
    }
    __syncthreads();
    wmma_gemm64<CIN, SRC_STRIDE, 80, XPS>(curbuf, sW, sXP);
    __syncthreads();
#pragma unroll 4
    for (int t = 0; t < 64; ++t) {
      float f0 = sXP[t * XPS + fd0] + sB[fd0];
      float f1 = sXP[t * XPS + fd0 + 1] + sB[fd0 + 1];
      float d  = sXP[t * XPS + 64 + a] + sB[64 + a];
      float pot = expf(-fabsf(d));
      float e0 = f0 * pot, e1 = f1 * pot;
      mx0 = fmaxf(mx0, e0);
      mx1 = fmaxf(mx1, e1);
      sm0 += e0;
      sm1 += e1;
    }
    __syncthreads();  // done reading sXP (and curbuf long before)
    cur ^= 1;
  }
  const size_t base = (size_t)b * 1024 + a * 128;
  pooled[base + fd0]          = mx0;
  pooled[base + fd0 + 1]      = mx1;
  pooled[base + 64 + fd0]     = sm0 * (1.f / TT);
  pooled[base + 64 + fd0 + 1] = sm1 * (1.f / TT);
}

// ---------------------------------------------------------------------------
// c_kernel: c[b][n] = b_out[n] + pooled[b] . w_out[cin: cin+1024, n]
// ---------------------------------------------------------------------------
__global__ __launch_bounds__(64) void c_kernel(
    const float* __restrict__ pooled,  // B x 1024
    const float* __restrict__ w_out,   // (cin+1024) x 64
    const float* __restrict__ b_out,   // 64
    float* __restrict__ cvec,          // B x 64
    int cin) {
  const int b = blockIdx.x, n = threadIdx.x;
  const float* w = w_out + (size_t)cin * 64;
  const float* p = pooled + (size_t)b * 1024;
  float acc = b_out[n];
  for (int i = 0; i < 1024; ++i) acc = fmaf(p[i], w[(size_t)i * 64 + n], acc);
  cvec[(size_t)b * 64 + n] = acc;
}

// ---------------------------------------------------------------------------
// h_kernel: h[b,t,:] = tanh(src[b,t,:] @ w_out[:CIN,:] + c[b,:])
// ---------------------------------------------------------------------------
template <int CIN>
__global__ __launch_bounds__(256) void h_kernel(
    const float* __restrict__ src,    // B x T x CIN
    const float* __restrict__ w_out,  // (CIN+1024) x 64 (head rows used)
    const float* __restrict__ cvec,   // B x 64
    float* __restrict__ h) {          // B x T x 64
  constexpr int SRC_STRIDE = CIN + 4;
  constexpr int OS = 68;
  __shared__ __align__(16) float sW[CIN * 64];
  __shared__ __align__(16) float sSrc[64 * SRC_STRIDE];
  __shared__ __align__(16) float sO[64 * OS];
  __shared__ float sC[64];

  const int b = blockIdx.x >> 5;  // T/64 = 32 chunks
  const int tc = (blockIdx.x & 31) * 64;
  const int tid = threadIdx.x;

  stage_async_issue<CIN, SRC_STRIDE>(src + ((size_t)b * TT + tc) * CIN, sSrc,
                                     tid);
  for (int i = tid; i < CIN * 64; i += 256) sW[i] = w_out[i];
  if (tid < 64) sC[tid] = cvec[(size_t)b * 64 + tid];
  wait_asynccnt<0>();
  __syncthreads();
  wmma_gemm64<CIN, SRC_STRIDE, 64, OS>(sSrc, sW, sO);
  __syncthreads();
  for (int i = tid; i < 64 * 64; i += 256) {
    int r = i >> 6, c = i & 63;
    h[((size_t)b * TT + tc + r) * 64 + c] = tanhf(sO[r * OS + c] + sC[c]);
  }
}

// ---------------------------------------------------------------------------
// final_kernel: h2 = tanh(h1 @ w_out2[:64,:] + c2); mean & first-argmax over
// T; then hh=[mean, argmax] -> relu(hh@w_p1+b_p1) @ w_p2 + b_p2.
// Double-buffered async LDS staging.
// ---------------------------------------------------------------------------
__global__ __launch_bounds__(256) void final_kernel(
    const float* __restrict__ h1,      // B x T x 64
    const float* __restrict__ w_out2,  // 1088 x 64 (head rows used)
    const float* __restrict__ c2,      // B x 64
    const float* __restrict__ w_p1,    // 128 x 64
    const float* __restrict__ b_p1,    // 64
    const float* __restrict__ w_p2,    // 64 x 2
    const float* __restrict__ b_p2,    // 2
    float* __restrict__ out) {         // B x 2
  constexpr int SRC_STRIDE = 68, OS = 68, OPW = 4;
  __shared__ __align__(16) float sW[64 * 64];
  __shared__ __align__(16) float sSrc0[64 * SRC_STRIDE];
  __shared__ __align__(16) float sSrc1[64 * SRC_STRIDE];
  __shared__ __align__(16) float sO[64 * OS];
  __shared__ float sC[64];
  __shared__ float red_sum[4][64];
  __shared__ float red_max[4][64];
  __shared__ int   red_idx[4][64];
  __shared__ float sHH[128];
  __shared__ float sP[64];

  const int b = blockIdx.x, tid = threadIdx.x;
  const float* gsrc = h1 + (size_t)b * TT * 64;

  stage_async_issue<64, SRC_STRIDE>(gsrc, sSrc0, tid);
  for (int i = tid; i < 64 * 64; i += 256) sW[i] = w_out2[i];
  if (tid < 64) sC[tid] = c2[(size_t)b * 64 + tid];

  const int fd = tid & 63;   // feature reduced by this thread
  const int grp = tid >> 6;  // stripe of 16 t's per chunk
  float msum = 0.f, mmax = -3.4e38f;
  int midx = 0;

  int cur = 0;
  for (int tc = 0; tc < TT; tc += 64) {
    const float* curbuf = cur ? sSrc1 : sSrc0;
    float* nxtbuf = cur ? sSrc0 : sSrc1;
    if (tc + 64 < TT) {
      stage_async_issue<64, SRC_STRIDE>(gsrc + (size_t)(tc + 64) * 64, nxtbuf,
                                        tid);
      wait_asynccnt<OPW>();
    } else {
      wait_asynccnt<0>();
    }
    __syncthreads();
    wmma_gemm64<64, SRC_STRIDE, 64, OS>(curbuf, sW, sO);
    __syncthreads();
#pragma unroll 4
    for (int k = 0; k < 16; ++k) {
      int t = grp * 16 + k;
      float v = tanhf(sO[t * OS + fd] + sC[fd]);
      msum += v;
      if (v > mmax) { mmax = v; midx = tc + t; }  // strict > keeps first idx
    }
    __syncthreads();
    cur ^= 1;
  }
  red_sum[grp][fd] = msum;
  red_max[grp][fd] = mmax;
  red_idx[grp][fd] = midx;
  __syncthreads();

  if (tid < 64) {
    float s = 0.f, mx = -3.4e38f;
    int ix = 0;
    for (int g = 0; g < 4; ++g) {
      s += red_sum[g][tid];
      float m = red_max[g][tid];
      int id = red_idx[g][tid];
      if (m > mx || (m == mx && id < ix)) { mx = m; ix = id; }
    }
    sHH[tid] = s * (1.f / TT);
    sHH[64 + tid] = (float)ix;
  }
  __syncthreads();
  if (tid < 64) {
    float acc = b_p1[tid];
    for (int i = 0; i < 128; ++i) acc = fmaf(sHH[i], w_p1[i * 64 + tid], acc);
    sP[tid] = fmaxf(acc, 0.f);
  }
  __syncthreads();
  if (tid < 2) {
    float acc = b_p2[tid];
    for (int i = 0; i < 64; ++i) acc = fmaf(sP[i], w_p2[i * 2 + tid], acc);
    out[(size_t)b * 2 + tid] = acc;
  }
}

// ---------------------------------------------------------------------------
extern "C" void kernel_launch(void* const* d_in, const int* in_sizes, int n_in,
                              void* d_out, int out_size, void* d_ws,
                              size_t ws_size, hipStream_t stream) {
  const float* x      = (const float*)d_in[0];
  const float* w_in1  = (const float*)d_in[1];
  const float* b_in1  = (const float*)d_in[2];
  const float* w_out1 = (const float*)d_in[3];
  const float* b_out1 = (const float*)d_in[4];
  const float* w_in2  = (const float*)d_in[5];
  const float* b_in2  = (const float*)d_in[6];
  const float* w_out2 = (const float*)d_in[7];
  const float* b_out2 = (const float*)d_in[8];
  const float* w_p1   = (const float*)d_in[9];
  const float* b_p1   = (const float*)d_in[10];
  const float* w_p2   = (const float*)d_in[11];
  const float* b_p2   = (const float*)d_in[12];
  float* out = (float*)d_out;

  // workspace layout (floats)
  float* ws = (float*)d_ws;
  float* pooled1 = ws;                   // 64*1024
  float* c1      = pooled1 + BB * 1024;  // 64*64
  float* pooled2 = c1 + BB * 64;         // 64*1024
  float* cc2     = pooled2 + BB * 1024;  // 64*64
  float* h1      = cc2 + BB * 64;        // 64*2048*64 (32 MB)

  (void)in_sizes; (void)n_in; (void)out_size; (void)ws_size;

  // Layer 1
  pool_kernel<NF><<<BB, 256, 0, stream>>>(x, w_in1, b_in1, pooled1);
  c_kernel<<<BB, 64, 0, stream>>>(pooled1, w_out1, b_out1, c1, NF);
  h_kernel<NF><<<BB * (TT / 64), 256, 0, stream>>>(x, w_out1, c1, h1);

  // Layer 2
  pool_kernel<FD><<<BB, 256, 0, stream>>>(h1, w_in2, b_in2, pooled2);
  c_kernel<<<BB, 64, 0, stream>>>(pooled2, w_out2, b_out2, cc2, FD);

  // Layer-2 output GEMM fused with mean/argmax reduction + MLP head
  final_kernel<<<BB, 256, 0, stream>>>(h1, w_out2, cc2, w_p1, b_p1, w_p2, b_p2,
                                       out);
}